// N3Tree_2456721293535
// MI455X (gfx1250) — compile-verified
//
#include <hip/hip_runtime.h>
#include <stdint.h>

// ---------------------------------------------------------------------------
// N3Tree query (N=2, DATA_DIM=16, MAX_STEPS=6) for MI455X / gfx1250.
//
// Roofline: mandatory HBM traffic = 24MB (points in) + 128MB (payload out)
// ~= 152MB -> ~6.5us @ 23.3 TB/s. child (1.2MB) and data (19MB) are
// L2-resident (192MB L2); payload reused ~7.6x per leaf so keep it RT-cached,
// stream the output with NT stores so it doesn't evict the hot tree.
// Zero FLOPs -> WMMA inapplicable; the CDNA5 feature path used here is the
// async global->LDS DMA (ASYNCcnt) for staging the top tree levels, plus
// wave32 branchless traversal.
// ---------------------------------------------------------------------------

#define TREE_N      2
#define DATA_DIM    16
#define MAX_STEPS   6          // DEPTH + 1
#define LDS_CELLS   4680       // cells of tree levels 0..3: (1+8+64+512)*8
#define BLOCK       256

#if defined(__has_builtin)
#  if __has_builtin(__builtin_amdgcn_global_load_async_to_lds_b128)
#    define HAVE_ASYNC_LDS 1
#  endif
#endif

typedef __attribute__((ext_vector_type(4))) float v4f;
typedef int v4i __attribute__((vector_size(16)));                 // int4
typedef __attribute__((address_space(1))) v4i* g_v4i_ptr;         // global
typedef __attribute__((address_space(3))) v4i* l_v4i_ptr;         // LDS

__global__ __launch_bounds__(BLOCK)
void n3tree_query_kernel(const float* __restrict__ pts,      // (Q,3)
                         const float* __restrict__ data,     // (cells, 16)
                         const int*   __restrict__ child,    // (cells,)
                         const float* __restrict__ invradius,// (3,)
                         const float* __restrict__ offset,   // (3,)
                         float* __restrict__ out,            // (Q,16)
                         int Q, int n_cells)
{
    __shared__ int s_child[LDS_CELLS];
    const int cached = (n_cells < LDS_CELLS) ? n_cells : LDS_CELLS;

    // ---- stage top-of-tree child pointers into LDS ----
#ifdef HAVE_ASYNC_LDS
    {
        const int chunks = cached >> 2;            // 16B chunks
        for (int i = (int)threadIdx.x; i < chunks; i += BLOCK) {
            __builtin_amdgcn_global_load_async_to_lds_b128(
                (g_v4i_ptr)(child + 4 * i),
                (l_v4i_ptr)(&s_child[4 * i]),
                /*imm offset*/ 0, /*cpol*/ 0);
        }
        // generic tail (empty for this tree: 4680 % 4 == 0)
        for (int i = (chunks << 2) + (int)threadIdx.x; i < cached; i += BLOCK)
            s_child[i] = child[i];
        asm volatile("s_wait_asynccnt 0" ::: "memory");
    }
#else
    for (int i = (int)threadIdx.x; i < cached; i += BLOCK)
        s_child[i] = child[i];
#endif
    __syncthreads();

    const int q = (int)blockIdx.x * BLOCK + (int)threadIdx.x;
    if (q >= Q) return;

    // ---- world -> tree coords, clamp to [0, 1] (1-1e-10 rounds to 1.0f) ----
    const float ir0 = invradius[0], ir1 = invradius[1], ir2 = invradius[2];
    const float of0 = offset[0],    of1 = offset[1],    of2 = offset[2];

    const float* p = pts + 3 * (size_t)q;
    float tx = __builtin_nontemporal_load(p + 0);
    float ty = __builtin_nontemporal_load(p + 1);
    float tz = __builtin_nontemporal_load(p + 2);
    tx = fminf(fmaxf(tx * ir0 + of0, 0.0f), 1.0f);
    ty = fminf(fmaxf(ty * ir1 + of1, 0.0f), 1.0f);
    tz = fminf(fmaxf(tz * ir2 + of2, 0.0f), 1.0f);

    // ---- branchless traversal; state is a fixed point once delta==0 ----
    int node = 0, leaf = 0;
#pragma unroll
    for (int s = 0; s < MAX_STEPS; ++s) {
        const float t2x = tx + tx, t2y = ty + ty, t2z = tz + tz;
        const int fx = (t2x >= 1.0f);        // == min(floor(t2), N-1) for N=2
        const int fy = (t2y >= 1.0f);
        const int fz = (t2z >= 1.0f);
        const int flat = node * 8 + (fx << 2) + (fy << 1) + fz;

        int delta;
        if (flat < cached) delta = s_child[flat];   // LDS: levels 0..3
        else               delta = child[flat];     // L2:  levels 4..5

        const bool term = (delta == 0);
        leaf = term ? flat : leaf;
        node += delta;                               // delta==0 -> unchanged
        tx = term ? tx : t2x - (float)fx;
        ty = term ? ty : t2y - (float)fy;
        tz = term ? tz : t2z - (float)fz;
    }

    // ---- gather 64B payload (L2-hot, RT) and stream it out (NT) ----
    const v4f* __restrict__ src =
        (const v4f*)(data + (size_t)leaf * DATA_DIM);
    v4f* dst = (v4f*)(out + (size_t)q * DATA_DIM);

    const v4f d0 = src[0];
    const v4f d1 = src[1];
    const v4f d2 = src[2];
    const v4f d3 = src[3];
    __builtin_nontemporal_store(d0, dst + 0);
    __builtin_nontemporal_store(d1, dst + 1);
    __builtin_nontemporal_store(d2, dst + 2);
    __builtin_nontemporal_store(d3, dst + 3);
}

extern "C" void kernel_launch(void* const* d_in, const int* in_sizes, int n_in,
                              void* d_out, int out_size, void* d_ws, size_t ws_size,
                              hipStream_t stream)
{
    (void)n_in; (void)out_size; (void)d_ws; (void)ws_size;

    const float* pts   = (const float*)d_in[0];   // indices  (Q*3)
    const float* data  = (const float*)d_in[1];   // data     (cells*16)
    const int*   child = (const int*)  d_in[2];   // child    (cells)
    const float* invr  = (const float*)d_in[3];   // invradius (3)
    const float* off   = (const float*)d_in[4];   // offset    (3)
    float*       out   = (float*)d_out;

    const int Q       = in_sizes[0] / 3;
    const int n_cells = in_sizes[2];

    const int grid = (Q + BLOCK - 1) / BLOCK;
    hipLaunchKernelGGL(n3tree_query_kernel, dim3(grid), dim3(BLOCK), 0, stream,
                       pts, data, child, invr, off, out, Q, n_cells);
}